// VectorQuantizer_18073222382323
// MI455X (gfx1250) — compile-verified
//
#include <hip/hip_runtime.h>

typedef __attribute__((ext_vector_type(2))) float v2f;
typedef __attribute__((ext_vector_type(8))) float v8f;

#define DIM        64
#define K_CODES    512
#define CHUNK      128          // codebook rows staged in LDS per phase
#define LDS_STRIDE 68           // padded floats per code row (272B: 16B-aligned, bank-spread)
#define NCHUNK     (K_CODES / CHUNK)
#define TILES_PER_CHUNK (CHUNK / 16)

// ---------------------------------------------------------------------------
// Kernel 1: codebook squared norms -> workspace
// ---------------------------------------------------------------------------
__global__ __launch_bounds__(256) void vq_cnorm_kernel(const float* __restrict__ cb,
                                                       float* __restrict__ cnorm) {
    int k = blockIdx.x * blockDim.x + threadIdx.x;
    if (k < K_CODES) {
        const float4* row = reinterpret_cast<const float4*>(cb + k * DIM);
        float s = 0.0f;
#pragma unroll
        for (int i = 0; i < DIM / 4; ++i) {
            float4 v = row[i];
            s += v.x * v.x + v.y * v.y + v.z * v.z + v.w * v.w;
        }
        cnorm[k] = s;
    }
}

// ---------------------------------------------------------------------------
// CDNA5 async global->LDS copy (ASYNCcnt-tracked, no VGPR round trip)
// ---------------------------------------------------------------------------
__device__ __forceinline__ void async_load_b128(unsigned lds_byte_off, const float* gaddr) {
    asm volatile("global_load_async_to_lds_b128 %0, %1, off"
                 :: "v"(lds_byte_off), "v"(gaddr)
                 : "memory");
}

__device__ __forceinline__ void wait_async_zero() {
    asm volatile("s_wait_asynccnt 0x0" ::: "memory");
}

// ---------------------------------------------------------------------------
// Kernel 2: WMMA cross term + running argmin.
// One wave32 handles 16 rows of x against all 512 codes.
// Codebook is double-buffered through LDS via async loads.
// ---------------------------------------------------------------------------
__global__ __launch_bounds__(256) void vq_argmin_kernel(const float* __restrict__ x,
                                                        const float* __restrict__ cb,
                                                        const float* __restrict__ cnorm,
                                                        int* __restrict__ out) {
    __shared__ float lds_cb[2][CHUNK * LDS_STRIDE];   // 2 x 34816 B

    const int tid  = threadIdx.x;
    const int wave = tid >> 5;          // 8 waves / block
    const int lane = tid & 31;
    const int ln   = lane & 15;         // position within 16-lane half
    const int g    = lane >> 4;         // half-wave select (K pair for A/B, row half for C/D)

    const int row_base = (blockIdx.x * 8 + wave) * 16;

    // --- Preload A fragments: full 16x64 x-tile for this wave ---------------
    // 32-bit A 16x4 layout: lanes 0-15 M=lane (K=0,1), lanes 16-31 M=lane-16 (K=2,3)
    v2f a[16];
    const v2f* xrow = reinterpret_cast<const v2f*>(x + (size_t)(row_base + ln) * DIM);
#pragma unroll
    for (int t = 0; t < 16; ++t) a[t] = xrow[2 * t + g];   // elements {4t+2g, 4t+2g+1}

    float bestv[8];
    int   besti[8];
#pragma unroll
    for (int i = 0; i < 8; ++i) { bestv[i] = 3.4e38f; besti[i] = 0; }

    // Generic->LDS pointer truncation: low 32 bits of the flat LDS aperture
    // address are the LDS byte offset (AS3 -> flat addrspacecast lowering).
    const unsigned lds_base0 = (unsigned)(uintptr_t)&lds_cb[0][0];
    const unsigned lds_base1 = (unsigned)(uintptr_t)&lds_cb[1][0];

    // Stage one chunk: 128 rows x 64 floats, 2048 b128 transfers / 256 threads.
    auto stage_async = [&](unsigned lds_base, int chunk) {
#pragma unroll
        for (int it = 0; it < (CHUNK * DIM / 4) / 256; ++it) {
            int li = tid + it * 256;           // float4 slot index
            int r  = li >> 4;                  // code row within chunk
            int c4 = li & 15;                  // float4 column
            async_load_b128(lds_base + (unsigned)(r * LDS_STRIDE + c4 * 4) * 4u,
                            cb + (size_t)(chunk * CHUNK + r) * DIM + c4 * 4);
        }
    };

    stage_async(lds_base0, 0);                 // prologue: fill buffer 0

    for (int c = 0; c < NCHUNK; ++c) {
        wait_async_zero();                     // my async stores to buf[c&1] landed
        __syncthreads();                       // all waves' staging + prior compute done
        if (c + 1 < NCHUNK)                    // prefetch next chunk into other buffer
            stage_async((c + 1) & 1 ? lds_base1 : lds_base0, c + 1);

        const float* buf = &lds_cb[c & 1][0];

#pragma unroll
        for (int tile = 0; tile < TILES_PER_CHUNK; ++tile) {
            const int code0 = c * CHUNK + tile * 16;
            const float csq = cnorm[code0 + ln];

            // B fragments (B = C^T, 4x16): b[t][j] = codebook[code0+ln][4t + 2g + j]
            v2f b[16];
            const v2f* crow = reinterpret_cast<const v2f*>(buf + (tile * 16 + ln) * LDS_STRIDE);
#pragma unroll
            for (int t = 0; t < 16; ++t) b[t] = crow[2 * t + g];

            // cross[16x16] = sum_k x[row][k] * c[code][k]
            v8f acc = {};
#pragma unroll
            for (int t = 0; t < 16; ++t) {
                acc = __builtin_amdgcn_wmma_f32_16x16x4_f32(
                    /*neg_a=*/false, a[t], /*neg_b=*/false, b[t],
                    /*c_mod=*/(short)0, acc, /*reuse_a=*/false, /*reuse_b=*/false);
            }

            // distance score (x_sq dropped: constant per row): c_sq - 2*cross
            const int idx = code0 + ln;
#pragma unroll
            for (int i = 0; i < 8; ++i) {
                float s = csq - 2.0f * acc[i];
                if (s < bestv[i]) { bestv[i] = s; besti[i] = idx; }
            }
        }
    }

    // --- argmin reduction across the 16 code columns (lanes within half) ----
#pragma unroll
    for (int off = 8; off >= 1; off >>= 1) {
#pragma unroll
        for (int i = 0; i < 8; ++i) {
            float ov = __shfl_xor(bestv[i], off, 32);
            int   oi = __shfl_xor(besti[i], off, 32);
            if (ov < bestv[i] || (ov == bestv[i] && oi < besti[i])) {
                bestv[i] = ov; besti[i] = oi;
            }
        }
    }

    // lane 0 holds rows 0-7, lane 16 holds rows 8-15
    if (ln == 0) {
#pragma unroll
        for (int i = 0; i < 8; ++i) out[row_base + g * 8 + i] = besti[i];
    }
}

// ---------------------------------------------------------------------------
extern "C" void kernel_launch(void* const* d_in, const int* in_sizes, int n_in,
                              void* d_out, int out_size, void* d_ws, size_t ws_size,
                              hipStream_t stream) {
    (void)n_in; (void)out_size; (void)ws_size;
    const float* x  = (const float*)d_in[0];
    const float* cb = (const float*)d_in[1];
    float* cnorm = (float*)d_ws;
    int*   out   = (int*)d_out;

    const int n_rows = in_sizes[0] / DIM;            // 524288
    const int blocks = n_rows / (8 * 16);            // 8 waves/block * 16 rows/wave

    vq_cnorm_kernel<<<(K_CODES + 255) / 256, 256, 0, stream>>>(cb, cnorm);
    vq_argmin_kernel<<<blocks, 256, 0, stream>>>(x, cb, cnorm, out);
}